// TemporalWeightedCrossEntropyLoss_32066225832748
// MI455X (gfx1250) — compile-verified
//
#include <hip/hip_runtime.h>
#include <hip/hip_bf16.h>

// Problem constants (match reference)
#define B_DIM   64
#define S_DIM   4096
#define C_DIM   128
#define WSZ     10
#define N_ROWS  (B_DIM * S_DIM)   // 262144
#define BLOCK   256
#define NBLK    (N_ROWS / BLOCK)  // 1024

typedef __attribute__((ext_vector_type(2))) float v2f;
typedef __attribute__((ext_vector_type(8))) float v8f;

// WINDOW[i] = linspace(5,1,11)[1:][i] = 4.6 - 0.4*i  (i in 0..9), WINDOW[9]=1.0

__global__ void __launch_bounds__(BLOCK)
twce_main_kernel(const float* __restrict__ x,
                 const int*   __restrict__ tgt,
                 float*       __restrict__ partial) {
    __shared__ float red[BLOCK];

    const int g = blockIdx.x * BLOCK + threadIdx.x;   // row index in [0, N_ROWS)
    float val = 0.0f;

    if (g < N_ROWS) {
        const int j       = g & (S_DIM - 1);  // position within sequence
        const int rowbase = g - j;            // start of this batch row in targets
        const int t       = tgt[g];

        // ---- temporal weight (faithful to reference index logic) ----
        float w = (t > 0) ? 5.0f : 1.0f;

        // left influence: event at e=j+o gives candidate WINDOW[WSZ-o] = 0.6 + 0.4*o
        #pragma unroll
        for (int o = 1; o <= WSZ; ++o) {
            if (j + o < S_DIM && tgt[rowbase + j + o] > 0)
                w = fmaxf(w, 0.6f + 0.4f * (float)o);
        }
        // right influence: event at e=j-k, L=min(WSZ,S-1-e), idx=L-k,
        // candidate WINDOW[idx] = 4.6 - 0.4*idx when idx >= 0
        #pragma unroll
        for (int k = 1; k <= WSZ; ++k) {
            const int e = j - k;
            if (e >= 0 && tgt[rowbase + e] > 0) {
                int L = S_DIM - 1 - e;
                if (L > WSZ) L = WSZ;
                const int idx = L - k;
                if (idx >= 0)
                    w = fmaxf(w, fmaf(-0.4f, (float)idx, 4.6f));
            }
        }

        // ---- log-softmax NLL over 128 classes (two streaming passes) ----
        const size_t  base = ((size_t)g) << 7;                 // g * 128
        const float4* row  = (const float4*)(x + base);        // 32 x b128 loads

        float m = -3.402823466e38f;
        #pragma unroll 8
        for (int i = 0; i < C_DIM / 4; ++i) {
            const float4 v = row[i];
            m = fmaxf(m, fmaxf(fmaxf(v.x, v.y), fmaxf(v.z, v.w)));
        }

        float s = 0.0f;
        #pragma unroll 8
        for (int i = 0; i < C_DIM / 4; ++i) {
            const float4 v = row[i];
            s += __expf(v.x - m) + __expf(v.y - m) +
                 __expf(v.z - m) + __expf(v.w - m);
        }

        const float xt  = x[base + (size_t)t];
        const float nll = -(xt - m - __logf(s));
        val = nll * w;
    }

    // ---- deterministic in-block tree reduction ----
    red[threadIdx.x] = val;
    __syncthreads();
    #pragma unroll
    for (int off = BLOCK / 2; off > 0; off >>= 1) {
        if ((int)threadIdx.x < off) red[threadIdx.x] += red[threadIdx.x + off];
        __syncthreads();
    }
    if (threadIdx.x == 0) partial[blockIdx.x] = red[0];
}

// Final reduction of NBLK=1024 partials by one full wave32 using
// V_WMMA_F32_16X16X4_F32 with an all-ones B matrix: D = A x 1 + C gives
// exact f32 row-sums; any bijection of 64 values onto A yields the total.
__global__ void __launch_bounds__(32)
twce_reduce_kernel(const float* __restrict__ partial,
                   float*       __restrict__ out) {
    const int lane = threadIdx.x;  // 0..31, EXEC all ones (WMMA requirement)

    v8f c = {0.f, 0.f, 0.f, 0.f, 0.f, 0.f, 0.f, 0.f};
    const v2f ones = {1.f, 1.f};

    // 16 chunks of 64 values: lane supplies A elements (2 f32 VGPRs per lane)
    for (int ch = 0; ch < NBLK / 64; ++ch) {
        v2f a;
        a.x = partial[ch * 64 + lane * 2 + 0];
        a.y = partial[ch * 64 + lane * 2 + 1];
        // 8 args: (neg_a, A, neg_b, B, c_mod, C, reuse_a, reuse_b)
        c = __builtin_amdgcn_wmma_f32_16x16x4_f32(
                false, a, false, ones, (short)0, c, false, false);
    }

    // lanes 0..15 hold rows M=0..7 in c[0..7]; lanes 16..31 hold M=8..15.
    float s = 0.0f;
    #pragma unroll
    for (int i = 0; i < 8; ++i) s += c[i];
    s += __shfl_xor(s, 16, 32);   // combine the two half-wave row groups

    if (lane == 0) out[0] = s * (1.0f / (float)N_ROWS);
}

extern "C" void kernel_launch(void* const* d_in, const int* in_sizes, int n_in,
                              void* d_out, int out_size, void* d_ws, size_t ws_size,
                              hipStream_t stream) {
    (void)in_sizes; (void)n_in; (void)out_size; (void)ws_size;

    const float* x   = (const float*)d_in[0];
    const int*   tgt = (const int*)d_in[1];
    float* partial   = (float*)d_ws;          // NBLK floats = 4 KB scratch
    float* outp      = (float*)d_out;         // 1 float

    twce_main_kernel<<<NBLK, BLOCK, 0, stream>>>(x, tgt, partial);
    twce_reduce_kernel<<<1, 32, 0, stream>>>(partial, outp);
}